// PagedOffloadTransformerBlock_54649163874777
// MI455X (gfx1250) — compile-verified
//
#include <hip/hip_runtime.h>
#include <math.h>

#define B_      64
#define DIM_    4096
#define H_      32
#define D_      128
#define BS_     16
#define MAXBLK_ 32
#define S_      512
#define EPS_    1e-5f

typedef float v2f __attribute__((ext_vector_type(2)));
typedef float v8f __attribute__((ext_vector_type(8)));

// ---------------------------------------------------------------------------
// CDNA5 async copy helpers (ASYNCcnt-tracked, bypass VGPRs)
// ---------------------------------------------------------------------------
__device__ __forceinline__ void async_cp_b128(const float* g, float* lds_generic) {
  // LDS destination address comes from the VDST VGPR; generic shared addr[31:0] == LDS offset
  unsigned loff = (unsigned)(size_t)lds_generic;
  asm volatile("global_load_async_to_lds_b128 %0, %1, off"
               :: "v"(loff), "v"(g) : "memory");
}
__device__ __forceinline__ void wait_asynccnt_le4() {
  asm volatile("s_wait_asynccnt 0x4" ::: "memory");
}

// ---------------------------------------------------------------------------
// RMSNorm: one block per row (64 rows), 256 threads
// ---------------------------------------------------------------------------
__global__ __launch_bounds__(256)
void rmsnorm_kernel(const float* __restrict__ x, const float* __restrict__ w,
                    float* __restrict__ out) {
  const int row = blockIdx.x;
  const float* xr = x + (size_t)row * DIM_;
  float* outr = out + (size_t)row * DIM_;
  __shared__ float red[256];
  float ss = 0.f;
  for (int i = threadIdx.x; i < DIM_; i += 256) { float v = xr[i]; ss += v * v; }
  red[threadIdx.x] = ss;
  __syncthreads();
  for (int off = 128; off > 0; off >>= 1) {
    if (threadIdx.x < off) red[threadIdx.x] += red[threadIdx.x + off];
    __syncthreads();
  }
  const float rs = rsqrtf(red[0] / (float)DIM_ + EPS_);
  for (int i = threadIdx.x; i < DIM_; i += 256) outr[i] = xr[i] * rs * w[i];
}

// ---------------------------------------------------------------------------
// GEMM: out[M=64,N] = A[64,K] * W[N,K]^T (+ res)   via V_WMMA_F32_16X16X4_F32
// Block = 256 threads = 8 wave32; block tile = 64x64.
// K staged through DOUBLE-BUFFERED LDS slabs (KC=32) filled by
// GLOBAL_LOAD_ASYNC_TO_LDS_B128: slab t+1 streams while WMMAs run on slab t.
// ---------------------------------------------------------------------------
#define KC   32
#define LDSP 36   // padded row stride (floats), 16B-aligned rows, conflict-free

__global__ __launch_bounds__(256)
void gemm_wmma_kernel(const float* __restrict__ A, const float* __restrict__ W,
                      const float* __restrict__ res, float* __restrict__ out,
                      int N, int K) {
  __shared__ float Alds[2][64 * LDSP];
  __shared__ float Blds[2][64 * LDSP];
  const int tid  = threadIdx.x;
  const int lane = tid & 31;
  const int wid  = tid >> 5;
  const int half = lane >> 4;     // 0: lanes 0-15, 1: lanes 16-31
  const int l    = lane & 15;
  const int wm   = wid & 3;       // M tile index (0..3)
  const int wn   = wid >> 2;      // N half (0..1)
  const int nblk = blockIdx.x * 64;

  // per-thread staging coordinates: 2 float4 chunks for A, 2 for W
  const int r0 = tid >> 3;                 // 0..31
  const int c0 = (tid & 7) << 2;           // 0,4,...,28
  const int r1 = (tid + 256) >> 3;         // 32..63
  const int c1 = c0;

  v8f acc0 = {};
  v8f acc1 = {};

  // prologue: slab 0 -> buffer 0 (4 async ops per thread)
  async_cp_b128(&A[(size_t)r0 * K + c0],            &Alds[0][r0 * LDSP + c0]);
  async_cp_b128(&A[(size_t)r1 * K + c1],            &Alds[0][r1 * LDSP + c1]);
  async_cp_b128(&W[(size_t)(nblk + r0) * K + c0],   &Blds[0][r0 * LDSP + c0]);
  async_cp_b128(&W[(size_t)(nblk + r1) * K + c1],   &Blds[0][r1 * LDSP + c1]);

  int buf = 0;
  for (int k0 = 0; k0 < K; k0 += KC) {
    // issue next slab into the other buffer (dummy re-load of slab 0 on the
    // final iteration keeps the wait immediate a constant: deterministic,
    // writes only into the dead buffer)
    const int kn = (k0 + KC < K) ? (k0 + KC) : 0;
    const int nb = buf ^ 1;
    async_cp_b128(&A[(size_t)r0 * K + kn + c0],          &Alds[nb][r0 * LDSP + c0]);
    async_cp_b128(&A[(size_t)r1 * K + kn + c1],          &Alds[nb][r1 * LDSP + c1]);
    async_cp_b128(&W[(size_t)(nblk + r0) * K + kn + c0], &Blds[nb][r0 * LDSP + c0]);
    async_cp_b128(&W[(size_t)(nblk + r1) * K + kn + c1], &Blds[nb][r1 * LDSP + c1]);

    // current slab's 4 ops (issued last iteration) complete in order:
    // <=4 outstanding means they have landed in LDS
    wait_asynccnt_le4();
    __syncthreads();

#pragma unroll
    for (int kk = 0; kk < KC; kk += 4) {
      // A fragment 16x4: lane half selects K pair {0,1} vs {2,3}
      v2f a;
      const int ai = (wm * 16 + l) * LDSP + kk + 2 * half;
      a.x = Alds[buf][ai];
      a.y = Alds[buf][ai + 1];
      // B fragments 4x16 (two N tiles): B[k][n] = W[n][k]
      v2f b0, b1;
      const int b0i = (wn * 32 + l) * LDSP + kk + 2 * half;
      b0.x = Blds[buf][b0i];
      b0.y = Blds[buf][b0i + 1];
      const int b1i = (wn * 32 + 16 + l) * LDSP + kk + 2 * half;
      b1.x = Blds[buf][b1i];
      b1.y = Blds[buf][b1i + 1];
      acc0 = __builtin_amdgcn_wmma_f32_16x16x4_f32(
          false, a, false, b0, (short)0, acc0, false, false);
      acc1 = __builtin_amdgcn_wmma_f32_16x16x4_f32(
          false, a, false, b1, (short)0, acc1, false, false);
    }
    __syncthreads();   // all waves done reading buf before it is refilled
    buf ^= 1;
  }

  // Epilogue: D VGPR r holds rows m = r (lanes 0-15) / r+8 (lanes 16-31)
#pragma unroll
  for (int r = 0; r < 8; ++r) {
    const int m  = wm * 16 + r + 8 * half;
    const int n0 = nblk + wn * 32 + l;
    const size_t i0 = (size_t)m * N + n0;
    const size_t i1 = i0 + 16;
    float v0 = acc0[r], v1 = acc1[r];
    if (res) { v0 += res[i0]; v1 += res[i1]; }
    out[i0] = v0;
    out[i1] = v1;
  }
}

// ---------------------------------------------------------------------------
// Paged attention (decode, q_len=1). One block per (b,h). The freshly
// projected k/v stand in for position s == ctx-1 (the slot_mapping slot),
// so the input heaps are never mutated.
// ---------------------------------------------------------------------------
__global__ __launch_bounds__(256)
void paged_attn_kernel(const float* __restrict__ q, const float* __restrict__ knew,
                       const float* __restrict__ vnew,
                       const float* __restrict__ key_heap,
                       const float* __restrict__ val_heap,
                       const int* __restrict__ block_table,
                       const int* __restrict__ ctx_lens,
                       const float* __restrict__ scale_p,
                       float* __restrict__ out) {
  const int b = blockIdx.x >> 5;   // / H_
  const int h = blockIdx.x & 31;   // % H_
  const int ctx = ctx_lens[b];
  const float scale = scale_p[0];
  const int tid = threadIdx.x;

  __shared__ float qs[D_];
  __shared__ float sc[S_];
  __shared__ float red[256];

  if (tid < D_) qs[tid] = q[(size_t)b * DIM_ + h * D_ + tid];
  __syncthreads();

  // scores (masked)
  for (int s = tid; s < S_; s += 256) {
    float val;
    if (s < ctx) {
      const float* kp;
      if (s == ctx - 1) {
        kp = &knew[(size_t)b * DIM_ + h * D_];
      } else {
        const int blk = block_table[b * MAXBLK_ + (s >> 4)];
        const int off = s & 15;
        kp = &key_heap[(((size_t)blk * BS_ + off) * H_ + h) * D_];
      }
      float dot = 0.f;
#pragma unroll 4
      for (int d = 0; d < D_; ++d) dot += qs[d] * kp[d];
      val = dot * scale;
    } else {
      val = -INFINITY;
    }
    sc[s] = val;
  }
  __syncthreads();

  // block max
  red[tid] = fmaxf(sc[tid], sc[tid + 256]);
  __syncthreads();
  for (int off = 128; off > 0; off >>= 1) {
    if (tid < off) red[tid] = fmaxf(red[tid], red[tid + off]);
    __syncthreads();
  }
  const float mx = red[0];
  __syncthreads();

  // exp + sum
  float ls = 0.f;
  for (int s = tid; s < S_; s += 256) {
    const float e = (s < ctx) ? __expf(sc[s] - mx) : 0.f;
    sc[s] = e;
    ls += e;
  }
  red[tid] = ls;
  __syncthreads();
  for (int off = 128; off > 0; off >>= 1) {
    if (tid < off) red[tid] += red[tid + off];
    __syncthreads();
  }
  const float inv = 1.f / red[0];
  __syncthreads();

  // weighted V accumulation: lanes 0..127 own one d each (coalesced rows)
  if (tid < D_) {
    float acc = 0.f;
    for (int s = 0; s < ctx; ++s) {
      float vv;
      if (s == ctx - 1) {
        vv = vnew[(size_t)b * DIM_ + h * D_ + tid];
      } else {
        const int blk = block_table[b * MAXBLK_ + (s >> 4)];
        const int off = s & 15;
        vv = val_heap[(((size_t)blk * BS_ + off) * H_ + h) * D_ + tid];
      }
      acc += sc[s] * vv;
    }
    out[(size_t)b * DIM_ + h * D_ + tid] = acc * inv;
  }
}

// ---------------------------------------------------------------------------
// SwiGLU elementwise: m = silu(g1) * g3
// ---------------------------------------------------------------------------
__global__ __launch_bounds__(256)
void swiglu_kernel(const float* __restrict__ g1, const float* __restrict__ g3,
                   float* __restrict__ out, int n) {
  const int i = blockIdx.x * 256 + threadIdx.x;
  if (i < n) {
    const float a = g1[i];
    out[i] = (a / (1.f + __expf(-a))) * g3[i];
  }
}

// ---------------------------------------------------------------------------
extern "C" void kernel_launch(void* const* d_in, const int* in_sizes, int n_in,
                              void* d_out, int out_size, void* d_ws, size_t ws_size,
                              hipStream_t stream) {
  (void)in_sizes; (void)n_in; (void)out_size; (void)ws_size;
  const float* x        = (const float*)d_in[0];
  const float* key_heap = (const float*)d_in[1];
  const float* val_heap = (const float*)d_in[2];
  const int*   btab     = (const int*)d_in[3];
  const int*   ctx      = (const int*)d_in[5];
  const float* scale_p  = (const float*)d_in[9];
  const float* wq = (const float*)d_in[13];
  const float* wk = (const float*)d_in[14];
  const float* wv = (const float*)d_in[15];
  const float* wo = (const float*)d_in[16];
  const float* n1 = (const float*)d_in[17];
  const float* n2 = (const float*)d_in[18];
  const float* w1 = (const float*)d_in[19];
  const float* w2 = (const float*)d_in[20];
  const float* w3 = (const float*)d_in[21];

  float* ws = (float*)d_ws;
  const size_t MD = (size_t)B_ * DIM_;      // 262144 floats
  const size_t MF = (size_t)B_ * 4 * DIM_;  // 1048576 floats
  float* h1   = ws;
  float* qb   = ws + 1 * MD;
  float* kn   = ws + 2 * MD;
  float* vn   = ws + 3 * MD;
  float* attn = ws + 4 * MD;
  float* x2   = ws + 5 * MD;
  float* h2   = ws + 6 * MD;
  float* g1   = ws + 7 * MD;
  float* g3   = g1 + MF;
  float* mb   = g3 + MF;

  // attention sub-block
  rmsnorm_kernel<<<B_, 256, 0, stream>>>(x, n1, h1);
  gemm_wmma_kernel<<<DIM_ / 64, 256, 0, stream>>>(h1, wq, nullptr, qb, DIM_, DIM_);
  gemm_wmma_kernel<<<DIM_ / 64, 256, 0, stream>>>(h1, wk, nullptr, kn, DIM_, DIM_);
  gemm_wmma_kernel<<<DIM_ / 64, 256, 0, stream>>>(h1, wv, nullptr, vn, DIM_, DIM_);
  paged_attn_kernel<<<B_ * H_, 256, 0, stream>>>(qb, kn, vn, key_heap, val_heap,
                                                 btab, ctx, scale_p, attn);
  gemm_wmma_kernel<<<DIM_ / 64, 256, 0, stream>>>(attn, wo, x, x2, DIM_, DIM_);

  // SwiGLU MLP sub-block
  rmsnorm_kernel<<<B_, 256, 0, stream>>>(x2, n2, h2);
  gemm_wmma_kernel<<<(4 * DIM_) / 64, 256, 0, stream>>>(h2, w1, nullptr, g1, 4 * DIM_, DIM_);
  gemm_wmma_kernel<<<(4 * DIM_) / 64, 256, 0, stream>>>(h2, w3, nullptr, g3, 4 * DIM_, DIM_);
  const int nsw = (int)(B_ * 4 * DIM_);
  swiglu_kernel<<<(nsw + 255) / 256, 256, 0, stream>>>(g1, g3, mb, nsw);
  gemm_wmma_kernel<<<DIM_ / 64, 256, 0, stream>>>(mb, w2, x2, (float*)d_out,
                                                  DIM_, 4 * DIM_);
}